// ProductionStorage_8134668059176
// MI455X (gfx1250) — compile-verified
//
#include <hip/hip_runtime.h>
#include <stdint.h>

// ---------------- problem constants ----------------
#define T_TOTAL   2097152            // rows
#define CHUNK     256                // rows per scan thread (exact part)
#define WARM      2048               // speculative warm-up steps (state contraction)
#define NCHUNK    (T_TOTAL / CHUNK)  // 8192 threads
#define TILE_ROWS 512                // rows staged to LDS per moment tile
#define TILE_FLOATS (TILE_ROWS * 6)  // 3072 floats = 12 KB
#define NTILES    (T_TOTAL / TILE_ROWS)
#define MOM_BLOCKS 64                // moment-kernel blocks (fixed-order reduction)

typedef float v2f __attribute__((ext_vector_type(2)));
typedef float v8f __attribute__((ext_vector_type(8)));

#if defined(__AMDGCN__) && __has_builtin(__builtin_amdgcn_wmma_f32_16x16x4_f32)
#define HAVE_WMMA_F32 1
#else
#define HAVE_WMMA_F32 0
#endif

#if defined(__AMDGCN__) && __has_builtin(__builtin_amdgcn_global_load_async_to_lds_b128) && \
    __has_builtin(__builtin_amdgcn_s_wait_asynccnt)
#define HAVE_ASYNC_LDS 1
#else
#define HAVE_ASYNC_LDS 0
#endif

// ============================================================================
// Kernel 1: chunked GR4J production-store scan.
// Each thread owns CHUNK rows; it warm-starts WARM steps earlier with s=0 to
// converge onto the true trajectory (contraction), then emits the 6 columns
// (P, E, p_n, e_n, p_s, perc) unnormalized into d_out.
// ============================================================================
__global__ __launch_bounds__(256) void gr4j_scan(const float2* __restrict__ x,
                                                 const float*  __restrict__ x1p,
                                                 float*        __restrict__ out) {
  const int chunk = blockIdx.x * 256 + threadIdx.x;
  if (chunk >= NCHUNK) return;

  const float X1    = x1p[0];
  const float invX1 = 1.0f / X1;
  const int   start = chunk * CHUNK;
  const int   end   = start + CHUNK;
  int i = start - WARM;
  if (i < 0) i = 0;                       // chunk 0 is exact (S_INIT == 0)

  float s = 0.0f;                          // S_INIT * x1
  for (; i < end; ++i) {
    const float2 pe = x[i];
    const float P  = pe.x, E = pe.y;
    const float pn = fmaxf(P - E, 0.0f);
    const float en = fmaxf(E - P, 0.0f);
    // tanh only of inputs -> off the serial dependency chain
    const float tp = tanhf(pn * invX1);
    const float te = tanhf(en * invX1);

    const float r  = s * invX1;
    const float ps = X1 * (1.0f - r * r) * tp / (1.0f + r * tp);
    const float es = s * (2.0f - r) * te / (1.0f + (1.0f - r) * te);
    s = s + ps - es;

    const float q    = (4.0f / 9.0f) * s * invX1;
    const float q2   = q * q;
    const float perc = s * (1.0f - rsqrtf(sqrtf(1.0f + q2 * q2))); // (1+q^4)^(-1/4)
    s = s - perc;

    if (i >= start) {
      float2* row = (float2*)(out + (size_t)i * 6);
      row[0] = make_float2(P, E);
      row[1] = make_float2(pn, en);
      row[2] = make_float2(ps, perc);
    }
  }
}

// ============================================================================
// Kernel 2: moment matrix D = [X|1]^T [X|1] via V_WMMA_F32_16X16X4_F32.
// Tiles of 512 rows x 6 cols are staged to LDS with async global->LDS copies.
// Per 4-row group: A(16x4) = Xaug^T chunk, B(4x16) = Xaug chunk; under the
// documented wave32 lane layouts both fragments are the identical 2 VGPRs.
// D[6][c] = column sums, D[c][c] = column sums of squares.
// Fixed-order (deterministic) reduction: wave -> LDS -> per-block partial.
// ============================================================================
#if HAVE_WMMA_F32
__global__ __launch_bounds__(256) void moment_kernel(const float* __restrict__ stage,
                                                     float*       __restrict__ part) {
  __shared__ float tile[TILE_FLOATS];     // 12 KB
  __shared__ float wacc[8][256];          // 8 KB, per-wave partial moment mats

  const int tid  = threadIdx.x;
  const int lane = tid & 31;
  const int wave = tid >> 5;              // 0..7
  const int c    = lane & 15;             // moment-matrix column this lane feeds
  const int k0   = (lane >> 4) * 2;       // K rows held by this lane (A layout)
  const int cc   = (c < 6) ? c : 0;       // clamped LDS column (avoid OOB)
  const float cval = (c == 6) ? 1.0f : 0.0f; // augmented "ones" column

  v8f acc = {0.f, 0.f, 0.f, 0.f, 0.f, 0.f, 0.f, 0.f};

  for (int t = blockIdx.x; t < NTILES; t += gridDim.x) {
    __syncthreads();                      // tile reuse barrier
    {
      const float4* s4 = (const float4*)(stage + (size_t)t * TILE_FLOATS);
      float4*       t4 = (float4*)tile;
#if HAVE_ASYNC_LDS
      // Builtin signature (from clang diagnostic):
      //   (int __vector(4) addrspace(1)* src, int __vector(4) addrspace(3)* lds,
      //    imm int offset, imm int cpol)
      typedef int v4i_g __attribute__((vector_size(16)));
      typedef __attribute__((address_space(1))) v4i_g* gptr_t;
      typedef __attribute__((address_space(3))) v4i_g* lptr_t;
#pragma unroll
      for (int j = 0; j < 3; ++j) {
        const int p = tid + j * 256;      // 768 float4s per tile
        __builtin_amdgcn_global_load_async_to_lds_b128(
            (gptr_t)(void*)(s4 + p), (lptr_t)(void*)(t4 + p), 0, 0);
      }
      __builtin_amdgcn_s_wait_asynccnt(0);
#else
#pragma unroll
      for (int j = 0; j < 3; ++j) {
        const int p = tid + j * 256;
        t4[p] = s4[p];
      }
#endif
    }
    __syncthreads();

    // 128 groups of 4 rows per tile; 16 groups per wave
#pragma unroll
    for (int j = 0; j < 16; ++j) {
      const int r0 = (wave * 16 + j) * 4;
      const float a0 = tile[(r0 + k0)     * 6 + cc];
      const float a1 = tile[(r0 + k0 + 1) * 6 + cc];
      v2f frag;
      frag.x = (c < 6) ? a0 : cval;
      frag.y = (c < 6) ? a1 : cval;
      // D += A x B  (A == B fragment-wise: symmetric moment accumulation)
      acc = __builtin_amdgcn_wmma_f32_16x16x4_f32(
          false, frag, false, frag, (short)0, acc, false, false);
    }
  }

  // deterministic in-block reduction of the 8 wave accumulators
  __syncthreads();
#pragma unroll
  for (int v = 0; v < 8; ++v) {
    const int m = v + 8 * (lane >> 4);    // C/D layout: VGPR v, lane half
    const int n = lane & 15;
    wacc[wave][m * 16 + n] = acc[v];
  }
  __syncthreads();
  float sum = 0.0f;
#pragma unroll
  for (int w = 0; w < 8; ++w) sum += wacc[w][tid];
  part[(size_t)blockIdx.x * 256 + tid] = sum;
}
#else
// Fallback (host parse / missing builtin): scalar column sums + sumsq written
// into the same moment-matrix slots (row 6 = sums, diagonal = sumsq).
__global__ __launch_bounds__(256) void moment_kernel(const float* __restrict__ stage,
                                                     float*       __restrict__ part) {
  __shared__ float fsum[256][12];
  const int tid = threadIdx.x;
  float sums[6] = {0, 0, 0, 0, 0, 0};
  float sq[6]   = {0, 0, 0, 0, 0, 0};
  for (size_t r = (size_t)blockIdx.x * 256 + tid; r < (size_t)T_TOTAL;
       r += (size_t)gridDim.x * 256) {
    const float* row = stage + r * 6;
#pragma unroll
    for (int cI = 0; cI < 6; ++cI) {
      const float v = row[cI];
      sums[cI] += v;
      sq[cI]   += v * v;
    }
  }
#pragma unroll
  for (int cI = 0; cI < 6; ++cI) { fsum[tid][cI] = sums[cI]; fsum[tid][6 + cI] = sq[cI]; }
  __syncthreads();
  float acc = 0.0f;
  if (tid < 12) {
    for (int j = 0; j < 256; ++j) acc += fsum[j][tid];
  }
  float outv = 0.0f;
  int slot = tid;
  if (tid < 12) {
    slot = (tid < 6) ? (6 * 16 + tid) : ((tid - 6) * 16 + (tid - 6));
    outv = acc;
  }
  part[(size_t)blockIdx.x * 256 + tid] = 0.0f;
  __syncthreads();
  if (tid < 12) part[(size_t)blockIdx.x * 256 + slot] = outv;
}
#endif

// ============================================================================
// Kernel 3: reduce block partials (fixed order) and derive mu / 1/sigma.
// ============================================================================
__global__ __launch_bounds__(256) void stats_kernel(const float* __restrict__ part,
                                                    float*       __restrict__ stat) {
  __shared__ float mom[256];
  const int tid = threadIdx.x;
  float sum = 0.0f;
  for (int b = 0; b < MOM_BLOCKS; ++b) sum += part[(size_t)b * 256 + tid];
  mom[tid] = sum;
  __syncthreads();
  if (tid < 6) {
    const float n   = (float)T_TOTAL;
    const float s1  = mom[6 * 16 + tid];      // column sum (ones row)
    const float s2  = mom[tid * 16 + tid];    // column sum of squares (diag)
    const float mu  = s1 / n;
    const float var = (s2 - s1 * mu) / (n - 1.0f);   // ddof = 1
    stat[tid]     = mu;
    stat[8 + tid] = rsqrtf(var);
  }
}

// ============================================================================
// Kernel 4: in-place z-score normalization, float2-vectorized.
// ============================================================================
__global__ __launch_bounds__(256) void norm_kernel(float* __restrict__ out,
                                                   const float* __restrict__ stat) {
  const int n2 = T_TOTAL * 3;                 // float2 elements
  float2* o2 = (float2*)out;
  for (int i = blockIdx.x * blockDim.x + threadIdx.x; i < n2;
       i += gridDim.x * blockDim.x) {
    const int p  = i % 3;                     // column pair {0,1},{2,3},{4,5}
    const int c0 = 2 * p;
    float2 v = o2[i];
    v.x = (v.x - stat[c0])     * stat[8 + c0];
    v.y = (v.y - stat[c0 + 1]) * stat[8 + c0 + 1];
    o2[i] = v;
  }
}

// ============================================================================
// Host launcher (graph-capture safe: only kernel launches on `stream`).
// ws layout: [0 .. 64*256) block moment partials | [16384 .. 16400) mu/rsigma.
// ============================================================================
extern "C" void kernel_launch(void* const* d_in, const int* in_sizes, int n_in,
                              void* d_out, int out_size, void* d_ws, size_t ws_size,
                              hipStream_t stream) {
  (void)in_sizes; (void)n_in; (void)out_size; (void)ws_size;
  const float2* x   = (const float2*)d_in[0];
  const float*  x1  = (const float*)d_in[1];
  float*        out = (float*)d_out;
  float*        part = (float*)d_ws;
  float*        stat = part + MOM_BLOCKS * 256;

  gr4j_scan   <<<NCHUNK / 256, 256, 0, stream>>>(x, x1, out);
  moment_kernel<<<MOM_BLOCKS,  256, 0, stream>>>(out, part);
  stats_kernel <<<1,           256, 0, stream>>>(part, stat);
  norm_kernel  <<<1024,        256, 0, stream>>>(out, stat);
}